// GCN_ancestor_sequential_84817014161572
// MI455X (gfx1250) — compile-verified
//
#include <hip/hip_runtime.h>
#include <math.h>

// ---------------- problem constants (match reference) ----------------
#define N_NODES 100000
#define F_IN    1024
#define HID     16
#define C_OUT   64
#define N_EDGES 3200000
#define BN_EPS  1e-5f

typedef __attribute__((ext_vector_type(2))) float v2f;
typedef __attribute__((ext_vector_type(8))) float v8f;

// ---------------- wave32 reductions ----------------
__device__ __forceinline__ float wave_max(float v) {
#pragma unroll
  for (int off = 16; off > 0; off >>= 1) v = fmaxf(v, __shfl_xor(v, off, 32));
  return v;
}
__device__ __forceinline__ float wave_sum(float v) {
#pragma unroll
  for (int off = 16; off > 0; off >>= 1) v += __shfl_xor(v, off, 32);
  return v;
}

// ---------------- degree / norm ----------------
__global__ void init_deg_kernel(float* __restrict__ deg) {
  int i = blockIdx.x * blockDim.x + threadIdx.x;
  if (i < N_NODES) deg[i] = 1.0f;  // self loop
}

__global__ void edge_deg_kernel(const int* __restrict__ dst, float* __restrict__ deg) {
  int e = blockIdx.x * blockDim.x + threadIdx.x;
  if (e < N_EDGES) atomicAdd(&deg[dst[e]], 1.0f);
}

__global__ void deg_to_dis_kernel(float* __restrict__ dis) {
  int i = blockIdx.x * blockDim.x + threadIdx.x;
  if (i < N_NODES) dis[i] = rsqrtf(fmaxf(dis[i], 1.0f));
}

__global__ void edge_norm_kernel(const int* __restrict__ src, const int* __restrict__ dst,
                                 const float* __restrict__ dis, float* __restrict__ enorm) {
  int e = blockIdx.x * blockDim.x + threadIdx.x;
  if (e < N_EDGES) enorm[e] = dis[src[e]] * dis[dst[e]];
}

// ---------------- GEMM1: h1[N,16] = x[N,1024] @ w1[1024,16] ----------------
// 8 waves/block; each wave computes two 16x16 tiles (32 rows) via f32 WMMA.
__global__ __launch_bounds__(256) void gemm1_kernel(const float* __restrict__ x,
                                                    const float* __restrict__ w1,
                                                    float* __restrict__ h1) {
  __shared__ __align__(16) float sB[F_IN * HID];  // 64 KB of 320 KB WGP LDS
  const int tid = threadIdx.x;
  {  // cooperative 128-bit loads of w1 into LDS
    const float4* w4 = (const float4*)w1;
    float4* s4 = (float4*)sB;
#pragma unroll
    for (int i = 0; i < (F_IN * HID / 4) / 256; ++i) s4[tid + i * 256] = w4[tid + i * 256];
  }
  __syncthreads();

  const int wave = tid >> 5;
  const int lane = tid & 31;
  const int half = lane >> 4;   // lane group 0/1
  const int mn   = lane & 15;   // M for A, N for B/C/D

  const int tile0 = (blockIdx.x * 8 + wave) * 2;
  if (tile0 >= N_NODES / 16) return;                 // wave-uniform
  const bool t1ok = (tile0 + 1) < (N_NODES / 16);    // wave-uniform
  const int row0 = tile0 * 16;
  const int row1 = row0 + 16;

  v8f acc0 = {}; v8f acc1 = {};
  for (int k = 0; k < F_IN; k += 4) {
    const int kk = k + 2 * half;  // K of .x element; .y is kk+1
    v2f b;
    b.x = sB[kk * HID + mn];
    b.y = sB[(kk + 1) * HID + mn];
    v2f a0 = *(const v2f*)(x + (size_t)(row0 + mn) * F_IN + kk);
    acc0 = __builtin_amdgcn_wmma_f32_16x16x4_f32(false, a0, false, b, (short)0, acc0, false, false);
    if (t1ok) {
      v2f a1 = *(const v2f*)(x + (size_t)(row1 + mn) * F_IN + kk);
      acc1 = __builtin_amdgcn_wmma_f32_16x16x4_f32(false, a1, false, b, (short)0, acc1, false, false);
    }
  }
#pragma unroll
  for (int v = 0; v < 8; ++v) {
    h1[(size_t)(row0 + v + 8 * half) * HID + mn] = acc0[v];
    if (t1ok) h1[(size_t)(row1 + v + 8 * half) * HID + mn] = acc1[v];
  }
}

// ---------------- edge aggregation, 16 channels ----------------
__global__ void agg16_kernel(const int* __restrict__ src, const int* __restrict__ dst,
                             const float* __restrict__ enorm, const float* __restrict__ h1,
                             float* __restrict__ agg1) {
  long tid = (long)blockIdx.x * blockDim.x + threadIdx.x;
  if (tid >= (long)N_EDGES * HID) return;
  int e = (int)(tid >> 4);
  int c = (int)(tid & 15);
  int s = src[e], d = dst[e];
  atomicAdd(&agg1[(size_t)d * HID + c], h1[(size_t)s * HID + c] * enorm[e]);
}

// a1 = relu(agg1 + h1*dis^2 + b1)   (in place on agg1)
__global__ void final1_kernel(float* __restrict__ agg1, const float* __restrict__ h1,
                              const float* __restrict__ dis, const float* __restrict__ b1) {
  int tid = blockIdx.x * blockDim.x + threadIdx.x;
  if (tid >= N_NODES * HID) return;
  int row = tid >> 4, c = tid & 15;
  float d = dis[row];
  float v = agg1[tid] + h1[tid] * d * d + b1[c];
  agg1[tid] = fmaxf(v, 0.0f);
}

// ---------------- GEMM2: h2[N,64] = a1[N,16] @ w2[16,64] ----------------
__global__ __launch_bounds__(256) void gemm2_kernel(const float* __restrict__ a1,
                                                    const float* __restrict__ w2,
                                                    float* __restrict__ h2) {
  __shared__ __align__(16) float sB[HID * C_OUT];  // 4 KB
  const int tid = threadIdx.x;
#pragma unroll
  for (int i = 0; i < (HID * C_OUT) / 256; ++i) sB[tid + i * 256] = w2[tid + i * 256];
  __syncthreads();

  const int wave = tid >> 5, lane = tid & 31;
  const int half = lane >> 4, mn = lane & 15;
  const int tile = blockIdx.x * 8 + wave;
  if (tile >= N_NODES / 16) return;
  const int row0 = tile * 16;

  v8f acc[4] = {};
  for (int k = 0; k < HID; k += 4) {
    const int kk = k + 2 * half;
    v2f a = *(const v2f*)(a1 + (size_t)(row0 + mn) * HID + kk);
#pragma unroll
    for (int nt = 0; nt < 4; ++nt) {
      v2f b;
      b.x = sB[kk * C_OUT + nt * 16 + mn];
      b.y = sB[(kk + 1) * C_OUT + nt * 16 + mn];
      acc[nt] = __builtin_amdgcn_wmma_f32_16x16x4_f32(false, a, false, b, (short)0, acc[nt], false, false);
    }
  }
#pragma unroll
  for (int nt = 0; nt < 4; ++nt)
#pragma unroll
    for (int v = 0; v < 8; ++v)
      h2[(size_t)(row0 + v + 8 * half) * C_OUT + nt * 16 + mn] = acc[nt][v];
}

// ---------------- edge aggregation, 64 channels ----------------
__global__ void agg64_kernel(const int* __restrict__ src, const int* __restrict__ dst,
                             const float* __restrict__ enorm, const float* __restrict__ h2,
                             float* __restrict__ agg2) {
  long tid = (long)blockIdx.x * blockDim.x + threadIdx.x;
  if (tid >= (long)N_EDGES * C_OUT) return;
  int e = (int)(tid >> 6);
  int c = (int)(tid & 63);
  int s = src[e], d = dst[e];
  atomicAdd(&agg2[(size_t)d * C_OUT + c], h2[(size_t)s * C_OUT + c] * enorm[e]);
}

// t = agg2 + h2*dis^2 + b2 ; agg2 := exp(weight_raw) * log_softmax(t)   (wave per row)
__global__ __launch_bounds__(256) void final2_kernel(float* __restrict__ agg2,
                                                     const float* __restrict__ h2,
                                                     const float* __restrict__ dis,
                                                     const float* __restrict__ b2,
                                                     const float* __restrict__ wraw) {
  const int wave = threadIdx.x >> 5, lane = threadIdx.x & 31;
  const int row = blockIdx.x * 8 + wave;
  if (row >= N_NODES) return;
  const float d = dis[row];
  const float selfn = d * d;
  const size_t base = (size_t)row * C_OUT;
  float t0 = agg2[base + lane]      + h2[base + lane]      * selfn + b2[lane];
  float t1 = agg2[base + lane + 32] + h2[base + lane + 32] * selfn + b2[lane + 32];
  float m  = wave_max(fmaxf(t0, t1));
  float se = wave_sum(expf(t0 - m) + expf(t1 - m));
  float lse = m + logf(se);
  float ew = expf(wraw[row]);
  agg2[base + lane]      = ew * (t0 - lse);
  agg2[base + lane + 32] = ew * (t1 - lse);
}

// ---------------- GEMM3: h4[N,64] = h3[N,64] @ mlp_w[64,64] + mlp_b ----------------
__global__ __launch_bounds__(256) void gemm3_kernel(const float* __restrict__ h3,
                                                    const float* __restrict__ mw,
                                                    const float* __restrict__ mb,
                                                    float* __restrict__ h4) {
  __shared__ __align__(16) float sB[C_OUT * C_OUT];  // 16 KB
  const int tid = threadIdx.x;
#pragma unroll
  for (int i = 0; i < (C_OUT * C_OUT) / 256; ++i) sB[tid + i * 256] = mw[tid + i * 256];
  __syncthreads();

  const int wave = tid >> 5, lane = tid & 31;
  const int half = lane >> 4, mn = lane & 15;
  const int tile = blockIdx.x * 8 + wave;
  if (tile >= N_NODES / 16) return;
  const int row0 = tile * 16;

  v8f acc[4] = {};
  for (int k = 0; k < C_OUT; k += 4) {
    const int kk = k + 2 * half;
    v2f a = *(const v2f*)(h3 + (size_t)(row0 + mn) * C_OUT + kk);
#pragma unroll
    for (int nt = 0; nt < 4; ++nt) {
      v2f b;
      b.x = sB[kk * C_OUT + nt * 16 + mn];
      b.y = sB[(kk + 1) * C_OUT + nt * 16 + mn];
      acc[nt] = __builtin_amdgcn_wmma_f32_16x16x4_f32(false, a, false, b, (short)0, acc[nt], false, false);
    }
  }
#pragma unroll
  for (int nt = 0; nt < 4; ++nt)
#pragma unroll
    for (int v = 0; v < 8; ++v)
      h4[(size_t)(row0 + v + 8 * half) * C_OUT + nt * 16 + mn] = acc[nt][v] + mb[nt * 16 + mn];
}

// ---------------- BatchNorm statistics ----------------
__global__ __launch_bounds__(256) void bn_stats_kernel(const float* __restrict__ h4,
                                                       float* __restrict__ sums) {
  __shared__ float ls[256], ls2[256];
  const int tid = threadIdx.x;
  const int c = tid & 63;
  const int sub = tid >> 6;  // 0..3
  float s = 0.f, s2 = 0.f;
  for (int row = blockIdx.x * 4 + sub; row < N_NODES; row += gridDim.x * 4) {
    float v = h4[(size_t)row * C_OUT + c];
    s += v; s2 += v * v;
  }
  ls[tid] = s; ls2[tid] = s2;
  __syncthreads();
  if (tid < 64) {
    s  = ls[tid]  + ls[tid + 64]  + ls[tid + 128]  + ls[tid + 192];
    s2 = ls2[tid] + ls2[tid + 64] + ls2[tid + 128] + ls2[tid + 192];
    atomicAdd(&sums[tid], s);
    atomicAdd(&sums[64 + tid], s2);
  }
}

// sums[c] -> mean ; sums[64+c] -> gamma * rsqrt(var+eps)
__global__ void bn_final_kernel(float* __restrict__ sums, const float* __restrict__ gamma) {
  const int c = threadIdx.x;
  if (c >= C_OUT) return;
  const float inv_n = 1.0f / (float)N_NODES;
  float mean = sums[c] * inv_n;
  float var = sums[64 + c] * inv_n - mean * mean;
  sums[c] = mean;
  sums[64 + c] = gamma[c] * rsqrtf(var + BN_EPS);
}

// out = log_softmax( (h4-mean)*scale + beta )    (wave per row)
__global__ __launch_bounds__(256) void final_out_kernel(const float* __restrict__ h4,
                                                        const float* __restrict__ stats,
                                                        const float* __restrict__ beta,
                                                        float* __restrict__ out) {
  const int wave = threadIdx.x >> 5, lane = threadIdx.x & 31;
  const int row = blockIdx.x * 8 + wave;
  if (row >= N_NODES) return;
  const size_t base = (size_t)row * C_OUT;
  float v0 = (h4[base + lane]      - stats[lane])      * stats[64 + lane]      + beta[lane];
  float v1 = (h4[base + lane + 32] - stats[lane + 32]) * stats[64 + lane + 32] + beta[lane + 32];
  float m  = wave_max(fmaxf(v0, v1));
  float se = wave_sum(expf(v0 - m) + expf(v1 - m));
  float lse = m + logf(se);
  out[base + lane]      = v0 - lse;
  out[base + lane + 32] = v1 - lse;
}

// ---------------- launcher ----------------
extern "C" void kernel_launch(void* const* d_in, const int* in_sizes, int n_in,
                              void* d_out, int out_size, void* d_ws, size_t ws_size,
                              hipStream_t stream) {
  const float* x     = (const float*)d_in[0];
  const int*   ei    = (const int*)d_in[1];
  const float* w1    = (const float*)d_in[2];
  const float* b1    = (const float*)d_in[3];
  const float* w2    = (const float*)d_in[4];
  const float* b2    = (const float*)d_in[5];
  const float* mlp_w = (const float*)d_in[6];
  const float* mlp_b = (const float*)d_in[7];
  const float* gamma = (const float*)d_in[8];
  const float* beta  = (const float*)d_in[9];
  const float* wraw  = (const float*)d_in[10];
  float* out = (float*)d_out;

  const int* src = ei;            // edge_index[0]
  const int* dst = ei + N_EDGES;  // edge_index[1]

  // workspace layout (floats)
  float* ws = (float*)d_ws;
  float* dis   = ws;                              // N
  float* enorm = dis + N_NODES;                   // E
  float* h1    = enorm + N_EDGES;                 // 16N
  float* agg1  = h1 + (size_t)N_NODES * HID;      // 16N
  float* h2    = agg1 + (size_t)N_NODES * HID;    // 64N  (reused as h4 after GEMM3)
  float* agg2  = h2 + (size_t)N_NODES * C_OUT;    // 64N  (becomes h3 in place)
  float* sums  = agg2 + (size_t)N_NODES * C_OUT;  // 128

  // zero accumulators (capture-safe memset nodes)
  hipMemsetAsync(agg1, 0, (size_t)N_NODES * HID * sizeof(float), stream);
  hipMemsetAsync(agg2, 0, (size_t)N_NODES * C_OUT * sizeof(float), stream);
  hipMemsetAsync(sums, 0, 128 * sizeof(float), stream);

  const int TB = 256;
  // 1) degrees -> dis -> per-edge norm
  init_deg_kernel<<<(N_NODES + TB - 1) / TB, TB, 0, stream>>>(dis);
  edge_deg_kernel<<<(N_EDGES + TB - 1) / TB, TB, 0, stream>>>(dst, dis);
  deg_to_dis_kernel<<<(N_NODES + TB - 1) / TB, TB, 0, stream>>>(dis);
  edge_norm_kernel<<<(N_EDGES + TB - 1) / TB, TB, 0, stream>>>(src, dst, dis, enorm);

  // 2) GCN layer 1
  {
    int waveTasks = (N_NODES / 16 + 1) / 2;           // 2 tiles per wave
    int blocks = (waveTasks + 7) / 8;                 // 8 waves per block
    gemm1_kernel<<<blocks, TB, 0, stream>>>(x, w1, h1);
  }
  {
    long total = (long)N_EDGES * HID;
    agg16_kernel<<<(int)((total + TB - 1) / TB), TB, 0, stream>>>(src, dst, enorm, h1, agg1);
  }
  final1_kernel<<<(N_NODES * HID + TB - 1) / TB, TB, 0, stream>>>(agg1, h1, dis, b1);

  // 3) GCN layer 2
  {
    int blocks = (N_NODES / 16 + 7) / 8;
    gemm2_kernel<<<blocks, TB, 0, stream>>>(agg1, w2, h2);
  }
  {
    long total = (long)N_EDGES * C_OUT;
    agg64_kernel<<<(int)((total + TB - 1) / TB), TB, 0, stream>>>(src, dst, enorm, h2, agg2);
  }
  final2_kernel<<<(N_NODES + 7) / 8, TB, 0, stream>>>(agg2, h2, dis, b2, wraw);

  // 4) MLP (h3 = agg2 in place) -> h4 into h2 buffer
  {
    int blocks = (N_NODES / 16 + 7) / 8;
    gemm3_kernel<<<blocks, TB, 0, stream>>>(agg2, mlp_w, mlp_b, h2);
  }

  // 5) BatchNorm + final log_softmax
  bn_stats_kernel<<<512, TB, 0, stream>>>(h2, sums);
  bn_final_kernel<<<1, 64, 0, stream>>>(sums, gamma);
  final_out_kernel<<<(N_NODES + 7) / 8, TB, 0, stream>>>(h2, sums, beta, out);

  (void)in_sizes; (void)n_in; (void)out_size; (void)ws_size;
}